// LogSqrt2Quantizer_59596966200119
// MI455X (gfx1250) — compile-verified
//
#include <hip/hip_runtime.h>
#include <stdint.h>

#define BLK   256
#define TILE  8192      // floats per TDM tile (32 KB); double buffered = 64 KB LDS
#define KMAX  33

#if defined(__has_builtin)
#if __has_builtin(__builtin_amdgcn_tensor_load_to_lds) && __has_builtin(__builtin_amdgcn_s_wait_tensorcnt)
#define HAVE_TDM 1
#endif
#endif

typedef float        vf4 __attribute__((ext_vector_type(4)));
typedef unsigned int v4u __attribute__((ext_vector_type(4)));
typedef int          v8i __attribute__((ext_vector_type(8)));
typedef int          v4i __attribute__((ext_vector_type(4)));

// k = 2*floor(log2(x_int)) + (frac>=0.5), x_int = round(x/s)+1, k in [0,32]
__device__ __forceinline__ int k_of(float xv, float s) {
  float q  = rintf(xv / s);          // jnp.round (RNE) of x_hat/s_x
  int   xi = (int)(q + 1.0f);        // +1 exact (q integral), second round is a no-op
  xi = xi < 1 ? 1 : xi;              // domain guard (reference requires x>=1)
  int li = 31 - __clz(xi);           // floor(log2)
  unsigned thr = 3u << li;           // frac bit: 2*xi >= 3*2^li
  int k = (li << 1) + ((((unsigned)xi << 1) >= thr) ? 1 : 0);
  return k > 32 ? 32 : k;
}

#if defined(HAVE_TDM)
// Build a 1-row TDM descriptor: copy TILE f32 from gsrc into LDS byte addr lds_byte.
// tensor_dim0 = remaining elements => tail reads beyond N are zero-filled by TDM OOB rule.
__device__ __forceinline__ void tdm_load_tile(const float* gsrc, unsigned lds_byte,
                                              long long remaining) {
  unsigned long long ga  = (unsigned long long)(size_t)gsrc;
  unsigned           td0 = (remaining >= 0x100000000ll) ? 0xFFFFFFFFu : (unsigned)remaining;
  v4u g0;
  g0[0] = 1u;                                            // count=1, user descriptor
  g0[1] = lds_byte;                                      // lds_addr (bytes)
  g0[2] = (unsigned)(ga & 0xFFFFFFFFull);                // global_addr[31:0]
  g0[3] = (unsigned)((ga >> 32) & 0x1FFFFFFull)          // global_addr[56:32]
        | (2u << 30);                                    // type=2 ("image")
  v8i g1;
  g1[0] = (int)(2u << 16);                               // data_size=4B; no wg mask/pad/iter
  g1[1] = (int)((td0 & 0xFFFFu) << 16);                  // tensor_dim0[15:0]
  g1[2] = (int)(((td0 >> 16) & 0xFFFFu) | (1u << 16));   // tensor_dim0[31:16]; tensor_dim1=1
  g1[3] = (int)((unsigned)TILE << 16);                   // tile_dim0 = TILE
  g1[4] = 1;                                             // tile_dim1=1, tile_dim2=0
  g1[5] = TILE;                                          // tensor_dim0_stride[31:0]
  g1[6] = 0;
  g1[7] = 0;
  v4i z4 = {0, 0, 0, 0};                                 // groups 2/3: dims 2..4 unused
  v8i z8 = {0, 0, 0, 0, 0, 0, 0, 0};                     // extra group (unused here)
  __builtin_amdgcn_tensor_load_to_lds(g0, g1, z4, z4, z8, 0);
}
#endif

__global__ void init_kernel(unsigned* __restrict__ presence) {
  presence[0] = 0u;
  presence[1] = 0u;
}

// Pass 1: stream x_hat, OR presence bits of k into 33-bit global mask,
// optionally spill k-bytes to workspace (L2-resident for pass 2).
__global__ void __launch_bounds__(BLK) pass1_kernel(
    const float* __restrict__ x, const float* __restrict__ sxp,
    unsigned* __restrict__ presence, unsigned char* __restrict__ kbuf,
    int use_k, long long N) {
  const float s  = sxp[0];
  unsigned mLo = 0u, mHi = 0u;
  const int tid = threadIdx.x;

#if defined(HAVE_TDM)
  __shared__ float tb[2][TILE];
  const long long numTiles = (N + TILE - 1) / (long long)TILE;
  const unsigned lds0 = (unsigned)(size_t)(&tb[0][0]);   // flat low 32 bits == LDS byte offset
  const unsigned lds1 = (unsigned)(size_t)(&tb[1][0]);
  const bool issuer = (tid < 32);                        // wave 0 owns TENSORcnt
  long long t0 = blockIdx.x;
  if (t0 < numTiles && issuer) tdm_load_tile(x + t0 * TILE, lds0, N - t0 * TILE);
  int buf = 0;
  for (long long tt = t0; tt < numTiles; tt += gridDim.x) {
    long long nxt  = tt + gridDim.x;
    bool      more = (nxt < numTiles);
    if (issuer) {
      if (more) {
        tdm_load_tile(x + nxt * TILE, buf ? lds0 : lds1, N - nxt * TILE);
        __builtin_amdgcn_s_wait_tensorcnt(1);            // older tile (tt) complete
      } else {
        __builtin_amdgcn_s_wait_tensorcnt(0);
      }
    }
    __syncthreads();                                     // tile tt visible to all waves
    const float*    tc   = &tb[buf][0];
    const long long base = tt * (long long)TILE;
    if (base + TILE <= N) {
      // Fast path: whole tile in-bounds -> no per-element checks
#pragma unroll
      for (int c = 0; c < TILE / (BLK * 4); ++c) {
        const int off = (c * BLK + tid) * 4;             // lane-contiguous b128 LDS reads
        vf4 v = *(const vf4*)(tc + off);
        unsigned kp = 0u;
#pragma unroll
        for (int j = 0; j < 4; ++j) {
          int k = k_of(v[j], s);
          if (k < 32) mLo |= (1u << k); else mHi |= 1u;
          kp |= ((unsigned)k) << (8 * j);
        }
        if (use_k) __builtin_nontemporal_store(kp, (unsigned*)(kbuf + base + off));
      }
    } else {
      // Tail tile (at most one per block): guarded path
#pragma unroll
      for (int c = 0; c < TILE / (BLK * 4); ++c) {
        const int off = (c * BLK + tid) * 4;
        vf4 v = *(const vf4*)(tc + off);
        long long g = base + off;
        unsigned kp = 0u;
        bool full = (g + 4 <= N);
#pragma unroll
        for (int j = 0; j < 4; ++j) {
          if (g + j < N) {
            int k = k_of(v[j], s);
            if (k < 32) mLo |= (1u << k); else mHi |= 1u;
            kp |= ((unsigned)k) << (8 * j);
          }
        }
        if (use_k) {
          if (full) {
            __builtin_nontemporal_store(kp, (unsigned*)(kbuf + g));
          } else {
            for (int j = 0; j < 4; ++j)
              if (g + j < N) kbuf[g + j] = (unsigned char)((kp >> (8 * j)) & 0xFFu);
          }
        }
      }
    }
    __syncthreads();                                     // all done reading tb[buf] before reuse
    buf ^= 1;
  }
#else
  // Fallback: plain vectorized non-temporal streaming loads
  const long long nq     = N >> 2;
  const long long stride = (long long)gridDim.x * BLK;
  for (long long i = (long long)blockIdx.x * BLK + tid; i < nq; i += stride) {
    vf4 v = __builtin_nontemporal_load((const vf4*)x + i);
    unsigned kp = 0u;
#pragma unroll
    for (int j = 0; j < 4; ++j) {
      int k = k_of(v[j], s);
      if (k < 32) mLo |= (1u << k); else mHi |= 1u;
      kp |= ((unsigned)k) << (8 * j);
    }
    if (use_k) __builtin_nontemporal_store(kp, (unsigned*)kbuf + i);
  }
  for (long long i = (nq << 2) + (long long)blockIdx.x * BLK + tid; i < N; i += stride) {
    int k = k_of(x[i], s);
    if (k < 32) mLo |= (1u << k); else mHi |= 1u;
    if (use_k) kbuf[i] = (unsigned char)k;
  }
#endif

  // wave32 OR-reduce, then one atomicOr per wave per word
  for (int o = 16; o > 0; o >>= 1) {
    mLo |= __shfl_xor(mLo, o, 32);
    mHi |= __shfl_xor(mHi, o, 32);
  }
  if ((tid & 31) == 0) {
    if (mLo) atomicOr(&presence[0], mLo);
    if (mHi) atomicOr(&presence[1], mHi);
  }
}

// Tiny scalar kernel: replicate the reference's 33-slot table construction exactly.
__global__ void table_kernel(const unsigned* __restrict__ presence,
                             const float* __restrict__ sxp,
                             float* __restrict__ table,
                             float* __restrict__ out_last) {
  const float s  = sxp[0];
  const float sn = s * 255.0f;
  *out_last = sn;                                        // second tuple output: s_x_new
  const unsigned pLo = presence[0];
  const unsigned pHi = presence[1];
  const float fbase = 1.0001f;
  const float c = rintf((1.0f / log2f(fbase)) * 0.1f);   // 693
  float vals[KMAX];
  bool  pres[KMAX];
  for (int g = 0; g < KMAX; ++g) {
    bool p = (g < 32) ? (((pLo >> g) & 1u) != 0u) : ((pHi & 1u) != 0u);
    pres[g] = p;
    float v10  = (float)(5 * g);
    float ylog = rintf(-v10 * c);                        // x_int_log_q per slot
    float dq   = rintf(powf(fbase, -ylog));              // round(base**(693*v10))
    float x8   = rintf((dq - 1.0f) / 255.0f);
    x8 = fminf(fmaxf(x8, 0.0f), 255.0f);
    vals[g] = p ? x8 : -1.0f;
  }
  for (int i = 1; i < KMAX; ++i) {                       // sort descending
    float v = vals[i]; int j = i - 1;
    while (j >= 0 && vals[j] < v) { vals[j + 1] = vals[j]; --j; }
    vals[j + 1] = v;
  }
  float int8_map[KMAX];
  for (int i = 0; i < KMAX; ++i) int8_map[i] = 0.0f;
  {
    int rank = -1;
    for (int j = 0; j < KMAX; ++j) {
      bool is_new = (j == 0) ? (vals[0] >= 0.0f)
                             : ((vals[j] != vals[j - 1]) && (vals[j] >= 0.0f));
      if (is_new) { ++rank; if (rank < KMAX) int8_map[rank] = vals[j]; }
      // duplicates/sentinels dumped (reference writes them into discarded slot K)
    }
  }
  int pv[KMAX];
  for (int g = 0; g < KMAX; ++g) pv[g] = pres[g] ? 5 * g : -1;
  for (int i = 1; i < KMAX; ++i) {                       // sort descending
    int v = pv[i]; int j = i - 1;
    while (j >= 0 && pv[j] < v) { pv[j + 1] = pv[j]; --j; }
    pv[j + 1] = v;
  }
  const int thr = pv[15];                                // 16th-largest present v10
  bool kept[KMAX]; int n_kept = 0;
  for (int g = 0; g < KMAX; ++g) {
    kept[g] = pres[g] && ((5 * g) > thr);
    n_kept += kept[g] ? 1 : 0;
  }
  int ngt[KMAX]; int acc = 0;
  for (int g = KMAX - 1; g >= 0; --g) { ngt[g] = acc; if (kept[g]) ++acc; }
  const int nk = n_kept > 32 ? 32 : n_kept;
  for (int g = 0; g < KMAX; ++g) {
    int idx = kept[g] ? (ngt[g] > 32 ? 32 : ngt[g]) : nk;
    table[g] = int8_map[idx] * sn;                       // pre-scale by s_x_new
  }
}

// Pass 2a: gather from L2-resident k-bytes (preferred; 101 MB fits 192 MB L2).
__global__ void __launch_bounds__(BLK) pass2_from_k(
    const unsigned char* __restrict__ kbuf, const float* __restrict__ table,
    float* __restrict__ out, long long N) {
  __shared__ float t[KMAX];
  if (threadIdx.x < KMAX) t[threadIdx.x] = table[threadIdx.x];
  __syncthreads();
  const long long nq     = N >> 2;
  const long long stride = (long long)gridDim.x * BLK;
  const unsigned* kw = (const unsigned*)kbuf;
  for (long long i = (long long)blockIdx.x * BLK + threadIdx.x; i < nq; i += stride) {
    __builtin_prefetch(kw + i + 8 * stride, 0, 1);       // global_prefetch_b8
    unsigned kk = kw[i];
    vf4 o;
    o[0] = t[kk & 0xFFu];
    o[1] = t[(kk >> 8) & 0xFFu];
    o[2] = t[(kk >> 16) & 0xFFu];
    o[3] = t[(kk >> 24)];
    __builtin_nontemporal_store(o, (vf4*)out + i);
  }
  for (long long i = (nq << 2) + (long long)blockIdx.x * BLK + threadIdx.x; i < N; i += stride)
    out[i] = t[kbuf[i]];
}

// Pass 2b: workspace too small -> recompute k from x_hat.
__global__ void __launch_bounds__(BLK) pass2_from_x(
    const float* __restrict__ x, const float* __restrict__ sxp,
    const float* __restrict__ table, float* __restrict__ out, long long N) {
  __shared__ float t[KMAX];
  if (threadIdx.x < KMAX) t[threadIdx.x] = table[threadIdx.x];
  __syncthreads();
  const float s = sxp[0];
  const long long nq     = N >> 2;
  const long long stride = (long long)gridDim.x * BLK;
  for (long long i = (long long)blockIdx.x * BLK + threadIdx.x; i < nq; i += stride) {
    vf4 v = __builtin_nontemporal_load((const vf4*)x + i);
    vf4 o;
#pragma unroll
    for (int j = 0; j < 4; ++j) o[j] = t[k_of(v[j], s)];
    __builtin_nontemporal_store(o, (vf4*)out + i);
  }
  for (long long i = (nq << 2) + (long long)blockIdx.x * BLK + threadIdx.x; i < N; i += stride)
    out[i] = t[k_of(x[i], s)];
}

extern "C" void kernel_launch(void* const* d_in, const int* in_sizes, int n_in,
                              void* d_out, int out_size, void* d_ws, size_t ws_size,
                              hipStream_t stream) {
  (void)n_in; (void)out_size;
  const float* x  = (const float*)d_in[0];
  const float* sx = (const float*)d_in[1];
  const long long N = (long long)in_sizes[0];

  // Workspace layout: [0,8) presence bits; [128,260) code table; [1024, 1024+N) k-bytes
  unsigned*      presence = (unsigned*)d_ws;
  float*         table    = (float*)((char*)d_ws + 128);
  unsigned char* kbuf     = (unsigned char*)((char*)d_ws + 1024);
  const int use_k = (ws_size >= (size_t)N + 1024) ? 1 : 0;
  float* out = (float*)d_out;

  init_kernel<<<1, 1, 0, stream>>>(presence);

  const long long numTiles = (N + TILE - 1) / (long long)TILE;
  int blocks1 = (int)(numTiles < 4096 ? numTiles : 4096);
  if (blocks1 < 1) blocks1 = 1;
  pass1_kernel<<<blocks1, BLK, 0, stream>>>(x, sx, presence, kbuf, use_k, N);

  table_kernel<<<1, 1, 0, stream>>>(presence, sx, table, out + N);

  const long long quads = (N + 3) >> 2;
  long long b2 = (quads + BLK - 1) / BLK;
  int blocks2 = (int)(b2 < 4096 ? b2 : 4096);
  if (blocks2 < 1) blocks2 = 1;
  if (use_k) pass2_from_k<<<blocks2, BLK, 0, stream>>>(kbuf, table, out, N);
  else       pass2_from_x<<<blocks2, BLK, 0, stream>>>(x, sx, table, out, N);
}